// QuantSoftmax_90443421319774
// MI455X (gfx1250) — compile-verified
//
#include <hip/hip_runtime.h>
#include <cstdint>
#include <cstddef>

// ---------------------------------------------------------------------------
// QuantSoftmax for MI455X (gfx1250, wave32).
// Memory-bound: ~1.2 GB HBM traffic -> ~52us floor @ 23.3 TB/s.
// Uses CDNA5 async global->LDS copies (ASYNCcnt / s_wait_asynccnt) to stage
// each 4KB row, wave32 shuffle reductions, one v_exp_f32 per element.
// (qsm_row_kernel is defined first so its asm shows in the disasm snippet.)
// ---------------------------------------------------------------------------

#define G_VALUE      3.0517578125e-05f   // 2^-15
#define I32MAX_F     2147483647.0f       // rounds to 2^31 in fp32, same as JAX
#define INV_65536_F  1.52587890625e-05f  // 2^-16 (exact)

typedef int v4i __attribute__((ext_vector_type(4)));
typedef v4i __attribute__((address_space(1))) *global_v4i_ptr;
typedef v4i __attribute__((address_space(3))) *lds_v4i_ptr;

#if __has_builtin(__builtin_amdgcn_global_load_async_to_lds_b128)
#define USE_ASYNC_LDS 1
#else
#define USE_ASYNC_LDS 0
#endif

#if __has_builtin(__builtin_amdgcn_s_wait_asynccnt)
#define WAIT_ASYNC_0() __builtin_amdgcn_s_wait_asynccnt(0)
#else
#define WAIT_ASYNC_0() asm volatile("s_wait_asynccnt 0" ::: "memory")
#endif

__device__ __forceinline__ float wave_reduce_max_f32(float v) {
#pragma unroll
  for (int off = 16; off > 0; off >>= 1)
    v = fmaxf(v, __shfl_xor(v, off, 32));
  return v;
}

__device__ __forceinline__ float wave_reduce_sum_f32(float v) {
#pragma unroll
  for (int off = 16; off > 0; off >>= 1)
    v += __shfl_xor(v, off, 32);
  return v;
}

// ----------------- kernel 2: per-row quantized softmax (1024 wide) ----------
__global__ __launch_bounds__(256) void qsm_row_kernel(
    const float* __restrict__ x, float* __restrict__ y,
    const unsigned int* __restrict__ amax_bits) {
  __shared__ __align__(16) float lds_row[1024];
  __shared__ float red_max[8];
  __shared__ float red_sum[8];

  const int t = threadIdx.x;
  const int lane = t & 31, wave = t >> 5;
  const size_t base = (size_t)blockIdx.x * 1024 + (size_t)t * 4;

  // ---- stage this thread's 16B of the row through LDS (CDNA5 async path) ---
  float4 v;
#if USE_ASYNC_LDS
  __builtin_amdgcn_global_load_async_to_lds_b128(
      (global_v4i_ptr)(x + base),
      (lds_v4i_ptr)(&lds_row[t * 4]),
      /*offset=*/0, /*cpol=*/0);
#endif

  // Per-tensor scale (scalar; real divides are fine here, once per block).
  const float amax      = __uint_as_float(*amax_bits);
  const float scale     = amax / 127.0f;
  const float inv_scale = 1.0f / scale;
  const float x0        = floorf(-inv_scale);   // <= -1
  const float inv_x0    = 1.0f / x0;
  const float lo        = 15.0f * x0;           // CONST_N * x0_int

#if USE_ASYNC_LDS
  WAIT_ASYNC_0();  // each lane reads back only its own 16B -> no barrier needed
  v = *reinterpret_cast<const float4*>(&lds_row[t * 4]);
#else
  v = *reinterpret_cast<const float4*>(x + base);
#endif

  // ---- symmetric int8 quantize (round half-even like jnp.round) ------------
  float qx = fminf(fmaxf(rintf(v.x * inv_scale), -128.0f), 127.0f);
  float qy = fminf(fmaxf(rintf(v.y * inv_scale), -128.0f), 127.0f);
  float qz = fminf(fmaxf(rintf(v.z * inv_scale), -128.0f), 127.0f);
  float qw = fminf(fmaxf(rintf(v.w * inv_scale), -128.0f), 127.0f);

  // ---- row max (1024 elems: wave shuffle + 8-way LDS combine) --------------
  float m = fmaxf(fmaxf(qx, qy), fmaxf(qz, qw));
  m = wave_reduce_max_f32(m);
  if (lane == 0) red_max[wave] = m;
  __syncthreads();
  float rm = red_max[0];
#pragma unroll
  for (int w = 1; w < 8; ++w) rm = fmaxf(rm, red_max[w]);

  // ---- integer-style exp via shifts (all fp32 floors, matches reference) ---
  auto iexp = [&](float xi) -> float {
    xi -= rm;                                               // in [-255, 0]
    float ti = xi + floorf(xi * 0.5f) - floorf(xi * 0.0625f);
    ti = fmaxf(ti, lo);
    float q = floorf(ti * inv_x0);                          // 0..15
    float r = ti - x0 * q;
    float ei = r * 0.5f - x0;
    return fmaxf(floorf(ei * exp2f(15.0f - q)), 0.0f);      // v_exp_f32
  };
  float ex = iexp(qx), ey = iexp(qy), ez = iexp(qz), ew = iexp(qw);

  // ---- row sum -------------------------------------------------------------
  float s = (ex + ey) + (ez + ew);
  s = wave_reduce_sum_f32(s);
  if (lane == 0) red_sum[wave] = s;
  __syncthreads();
  float rs = 0.0f;
#pragma unroll
  for (int w = 0; w < 8; ++w) rs += red_sum[w];
  rs = fminf(rs, I32MAX_F);
  const float factor = floorf(I32MAX_F / rs);

  // ---- normalize & store ---------------------------------------------------
  float4 o;
  o.x = floorf(ex * factor * INV_65536_F) * G_VALUE;
  o.y = floorf(ey * factor * INV_65536_F) * G_VALUE;
  o.z = floorf(ez * factor * INV_65536_F) * G_VALUE;
  o.w = floorf(ew * factor * INV_65536_F) * G_VALUE;
  *reinterpret_cast<float4*>(y + base) = o;
}

// --------------------------- kernel 1: global |x| max -----------------------
__global__ __launch_bounds__(256) void qsm_absmax_kernel(
    const float4* __restrict__ x4, unsigned int* __restrict__ bits, int n4) {
  int i = blockIdx.x * 256 + threadIdx.x;
  const int stride = gridDim.x * 256;
  float m = 0.0f;
  for (; i < n4; i += stride) {
    __builtin_prefetch(x4 + i + stride, 0, 0);  // global_prefetch_b8 (speculative)
    float4 v = x4[i];
    m = fmaxf(m, fmaxf(fmaxf(fabsf(v.x), fabsf(v.y)),
                       fmaxf(fabsf(v.z), fabsf(v.w))));
  }
  m = wave_reduce_max_f32(m);
  __shared__ float red[8];
  const int lane = threadIdx.x & 31, wave = threadIdx.x >> 5;
  if (lane == 0) red[wave] = m;
  __syncthreads();
  if (threadIdx.x == 0) {
    float bm = red[0];
#pragma unroll
    for (int w = 1; w < 8; ++w) bm = fmaxf(bm, red[w]);
    // |x| >= 0 so IEEE bit pattern is monotone under unsigned compare.
    atomicMax(bits, __float_as_uint(bm));
  }
}

// --------------------------- kernel 0: init scratch -------------------------
__global__ void qsm_init_kernel(unsigned int* bits) {
  if (threadIdx.x == 0 && blockIdx.x == 0) *bits = 0u;
}

// ---------------------------------------------------------------------------
extern "C" void kernel_launch(void* const* d_in, const int* in_sizes, int n_in,
                              void* d_out, int out_size, void* d_ws, size_t ws_size,
                              hipStream_t stream) {
  const float* x = (const float*)d_in[0];
  float* out = (float*)d_out;
  unsigned int* bits = (unsigned int*)d_ws;

  const int n = in_sizes[0];      // 8*12*1024*1024 = 100,663,296
  const int n4 = n >> 2;          // float4 count
  const int rows = n >> 10;       // 98,304 rows of 1024

  qsm_init_kernel<<<1, 1, 0, stream>>>(bits);

  int blocks1 = 8192;             // enough waves to saturate 23.3 TB/s
  if (blocks1 * 256 > n4) blocks1 = (n4 + 255) / 256;
  qsm_absmax_kernel<<<blocks1, 256, 0, stream>>>((const float4*)x, bits, n4);

  qsm_row_kernel<<<rows, 256, 0, stream>>>(x, out, bits);
}